// APPNP_38981123178700
// MI455X (gfx1250) — compile-verified
//
#include <hip/hip_runtime.h>

typedef __attribute__((ext_vector_type(2))) float v2f;
typedef __attribute__((ext_vector_type(8))) float v8f;

// ---------------------------------------------------------------------------
// GEMM: C[M x N] = A[M x K] @ B[K x N], all fp32 row-major.
// One wave per 16x16 C tile, V_WMMA_F32_16X16X4_F32, K stepped by 4.
// A-frag (16x4): lane l<16 holds A[l][k0+0..1], lane l+16 holds A[l][k0+2..3].
// B-frag (4x16): v0 = rows k0 (low lanes) / k0+2 (high), v1 = k0+1 / k0+3.
// C/D: VGPR i = row (rowBase + 8*half + i), col = colBase + (lane&15).
// ---------------------------------------------------------------------------
__global__ __launch_bounds__(256)
void gemm_wmma_f32_kernel(const float* __restrict__ A, const float* __restrict__ B,
                          float* __restrict__ C, int M, int K, int N) {
  const int wave = threadIdx.x >> 5;
  const int lane = threadIdx.x & 31;
  const int half = lane >> 4;
  const int l    = lane & 15;
  const int rowBase = blockIdx.x << 4;
  const int colBase = wave << 4;
  if (rowBase >= M || colBase >= N) return;  // wave-uniform

  v8f acc = {};
  const float* arow = A + (size_t)(rowBase + l) * K;
  const float* bcol = B + (size_t)(colBase + l);
  for (int k0 = 0; k0 < K; k0 += 4) {
    const int ka = k0 + (half << 1);
    v2f a, b;
    a.x = arow[ka];
    a.y = arow[ka + 1];
    b.x = bcol[(size_t)ka * N];
    b.y = bcol[(size_t)(ka + 1) * N];
    acc = __builtin_amdgcn_wmma_f32_16x16x4_f32(
        /*neg_a=*/false, a, /*neg_b=*/false, b,
        /*c_mod=*/(short)0, acc, /*reuse_a=*/false, /*reuse_b=*/false);
  }
#pragma unroll
  for (int i = 0; i < 8; ++i) {
    C[(size_t)(rowBase + (half << 3) + i) * N + colBase + l] = acc[i];
  }
}

// ---------------------------------------------------------------------------
// Edge-weighted scatter-add: out[dst[e]*D + j] += scale * w[e] * x[src[e]*D + j]
// D = 1<<logd; D consecutive threads cover one edge -> coalesced gather +
// contiguous hardware f32 atomics (L2-resident working set on MI455X).
// ---------------------------------------------------------------------------
__global__ __launch_bounds__(256)
void scatter_add_kernel(const int* __restrict__ src, const int* __restrict__ dst,
                        const float* __restrict__ w, const float* __restrict__ x,
                        float* __restrict__ out, int nE, int logd, float scale) {
  const long long total  = (long long)nE << logd;
  const int dm1 = (1 << logd) - 1;
  const long long stride = (long long)gridDim.x * blockDim.x;
  for (long long idx = (long long)blockIdx.x * blockDim.x + threadIdx.x;
       idx < total; idx += stride) {
    const int e = (int)(idx >> logd);
    const int j = ((int)idx) & dm1;
    const int s = src[e];
    const int t = dst[e];
    const float v = scale * w[e] * x[((size_t)s << logd) + j];
    unsafeAtomicAdd(&out[((size_t)t << logd) + j], v);
  }
}

// out[i*D + j] = b[j]   (accumulator init with bias broadcast)
__global__ __launch_bounds__(256)
void init_bias_kernel(float* __restrict__ out, const float* __restrict__ b,
                      long long total, int dm1) {
  const long long stride = (long long)gridDim.x * blockDim.x;
  for (long long idx = (long long)blockIdx.x * blockDim.x + threadIdx.x;
       idx < total; idx += stride) {
    out[idx] = b[((int)idx) & dm1];
  }
}

// out[i] = alpha * h[i]   (APPNP teleport term init)
__global__ __launch_bounds__(256)
void init_scaled_kernel(float* __restrict__ out, const float* __restrict__ h,
                        float alpha, long long total) {
  const long long stride = (long long)gridDim.x * blockDim.x;
  for (long long idx = (long long)blockIdx.x * blockDim.x + threadIdx.x;
       idx < total; idx += stride) {
    out[idx] = alpha * h[idx];
  }
}

// trailing tuple scalar(s): reference returns (x, 10)
__global__ void tail_const_kernel(float* __restrict__ out, long long start,
                                  int n, float val) {
  int i = blockIdx.x * blockDim.x + threadIdx.x;
  if (i < n) out[start + i] = val;
}

extern "C" void kernel_launch(void* const* d_in, const int* in_sizes, int n_in,
                              void* d_out, int out_size, void* d_ws, size_t ws_size,
                              hipStream_t stream) {
  const float* features = (const float*)d_in[0];
  const int*   eidx     = (const int*)d_in[1];   // [2, nE] row-major
  const float* ew       = (const float*)d_in[2];
  const float* W1       = (const float*)d_in[3];
  const float* b1       = (const float*)d_in[4];
  const float* W2       = (const float*)d_in[5];
  const float* b2       = (const float*)d_in[6];

  const int IN_DIM = 256, H_DIM = 128, OUT_DIM = 64;
  const int nNodes = in_sizes[0] / IN_DIM;   // 50000
  const int nE     = in_sizes[2];            // 800000
  const int* src = eidx;
  const int* dst = eidx + nE;

  float* buf0 = (float*)d_ws;                         // nNodes*128 floats
  float* buf1 = buf0 + (size_t)nNodes * H_DIM;        // nNodes*128 floats

  const long long tot128 = (long long)nNodes * H_DIM;
  const long long tot64  = (long long)nNodes * OUT_DIM;
  dim3 blk(256);
  const int SC_BLOCKS = 4096, EL_BLOCKS = 2048;
  const int mTiles = (nNodes + 15) / 16;

  // 1) y1 = X @ W1 -> buf0
  gemm_wmma_f32_kernel<<<dim3(mTiles), dim3(32 * (H_DIM / 16)), 0, stream>>>(
      features, W1, buf0, nNodes, IN_DIM, H_DIM);

  // 2) x1 = prop(y1) + b1 -> buf1
  init_bias_kernel<<<EL_BLOCKS, blk, 0, stream>>>(buf1, b1, tot128, H_DIM - 1);
  scatter_add_kernel<<<SC_BLOCKS, blk, 0, stream>>>(src, dst, ew, buf0, buf1,
                                                    nE, 7, 1.0f);

  // 3) y2 = x1 @ W2 -> buf0 (y1 dead)
  gemm_wmma_f32_kernel<<<dim3(mTiles), dim3(32 * (OUT_DIM / 16)), 0, stream>>>(
      buf1, W2, buf0, nNodes, H_DIM, OUT_DIM);

  // 4) h = prop(y2) + b2 -> buf1 (x1 dead after GEMM2 completes in-stream)
  init_bias_kernel<<<EL_BLOCKS, blk, 0, stream>>>(buf1, b2, tot64, OUT_DIM - 1);
  scatter_add_kernel<<<SC_BLOCKS, blk, 0, stream>>>(src, dst, ew, buf0, buf1,
                                                    nE, 6, 1.0f);

  // 5) APPNP: x_{t+1} = 0.9*prop(x_t) + 0.1*h, ping-pong in buf0 halves,
  //    final step writes directly to d_out.
  float* h  = buf1;
  float* p0 = buf0;
  float* p1 = buf0 + tot64;
  const float* cur = h;
  for (int t = 0; t < 10; ++t) {
    float* next = (t == 9) ? (float*)d_out : ((t & 1) ? p1 : p0);
    init_scaled_kernel<<<EL_BLOCKS, blk, 0, stream>>>(next, h, 0.1f, tot64);
    scatter_add_kernel<<<SC_BLOCKS, blk, 0, stream>>>(src, dst, ew, cur, next,
                                                      nE, 6, 0.9f);
    cur = next;
  }

  // 6) trailing scalar output (tuple second element == 10)
  if ((long long)out_size > tot64) {
    int extra = (int)((long long)out_size - tot64);
    tail_const_kernel<<<(extra + 255) / 256, blk, 0, stream>>>(
        (float*)d_out, tot64, extra, 10.0f);
  }
}